// SpatialMambaContextModel_51067161150074
// MI455X (gfx1250) — compile-verified
//
#include <hip/hip_runtime.h>
#include <hip/hip_bf16.h>
#include <math.h>

typedef __attribute__((ext_vector_type(2))) float v2f;
typedef __attribute__((ext_vector_type(8))) float v8f;

#define LN_EPSF 1e-5f
#define BN_EPSF 1e-5f

struct BranchP {
  const float* ln_g;   // (6,64)
  const float* ln_b;   // (6,64)
  const float* in_w;   // (6,64,256)
  const float* conv_w; // (6,128,4)
  const float* conv_b; // (6,128)
  const float* xproj_w;// (6,128,132)
  const float* dt_w;   // (6,4,128)
  const float* dt_b;   // (6,128)
  const float* Dp;     // (6,128)
  const float* out_w;  // (6,128,64)
};

__device__ __forceinline__ float sigm_f(float x) { return 1.f / (1.f + expf(-x)); }
__device__ __forceinline__ float silu_f(float x) { return x * sigm_f(x); }
__device__ __forceinline__ float softplus_f(float x) { return (x > 30.f) ? x : log1pf(expf(x)); }

// ---------------------------------------------------------------------------
// Kernel 1: per-branch single-token Mamba pipeline.
// Output at seq position 0 depends only on ctx + pos[0] (causality), so each
// branch is a 6-layer chain of tiny matvecs on one 64-vector.
// grid = 2 blocks (0 = topo, 1 = fuel), 256 threads each.
// ---------------------------------------------------------------------------
__global__ void __launch_bounds__(256)
branch_kernel(const float* __restrict__ ctx, const float* __restrict__ pos,
              BranchP p0, BranchP p1, float* __restrict__ tf_out)
{
  const BranchP p = (blockIdx.x == 0) ? p0 : p1;
  __shared__ float v[64];     // residual stream
  __shared__ float u[64];     // layernormed
  __shared__ float xz[256];
  __shared__ float xi[128];   // silu(conv(x)) at t=0
  __shared__ float sz[128];   // silu(z)
  __shared__ float xdbl[132]; // dt(4) | B(64) | C(64)
  __shared__ float yv[128];

  const int t = threadIdx.x;
  if (t < 64) v[t] = ctx[t] + pos[t];   // pos row 0
  __syncthreads();

  for (int i = 0; i < 6; ++i) {
    // ---- layernorm (all threads redo the tiny reduction via LDS broadcast)
    float mean = 0.f;
    for (int d = 0; d < 64; ++d) mean += v[d];
    mean *= (1.f / 64.f);
    float var = 0.f;
    for (int d = 0; d < 64; ++d) { float c = v[d] - mean; var += c * c; }
    var *= (1.f / 64.f);
    const float rstd = rsqrtf(var + LN_EPSF);
    if (t < 64) u[t] = (v[t] - mean) * rstd * p.ln_g[i * 64 + t] + p.ln_b[i * 64 + t];
    __syncthreads();

    // ---- xz[e] = u . in_w[i][:, e]   (e = 0..255)
    {
      float acc = 0.f;
      const float* W = p.in_w + i * 64 * 256 + t;
      for (int d = 0; d < 64; ++d) acc += u[d] * W[d * 256];
      xz[t] = acc;
    }
    __syncthreads();

    // ---- split, causal conv @ t=0 (only tap K-1=3), silu
    if (t < 128) {
      const float xiv = xz[t];
      const float zv  = xz[128 + t];
      const float c = p.conv_b[i * 128 + t] + xiv * p.conv_w[(i * 128 + t) * 4 + 3];
      xi[t] = silu_f(c);
      sz[t] = silu_f(zv);
    }
    __syncthreads();

    // ---- xdbl[e] = xi . xproj_w[i][:, e]   (e = 0..131)
    if (t < 132) {
      float acc = 0.f;
      const float* W = p.xproj_w + i * 128 * 132 + t;
      for (int j = 0; j < 128; ++j) acc += xi[j] * W[j * 132];
      xdbl[t] = acc;
    }
    __syncthreads();

    // ---- delta, scan at t=0: y = xi*(delta*(B.C) + D), then * silu(z)
    if (t < 128) {
      float dtv = p.dt_b[i * 128 + t];
      for (int r = 0; r < 4; ++r) dtv += xdbl[r] * p.dt_w[(i * 4 + r) * 128 + t];
      const float delta = softplus_f(dtv);
      float bc = 0.f;
      for (int n = 0; n < 64; ++n) bc += xdbl[4 + n] * xdbl[68 + n];
      yv[t] = xi[t] * (delta * bc + p.Dp[i * 128 + t]) * sz[t];
    }
    __syncthreads();

    // ---- residual: v += y @ out_w[i]
    if (t < 64) {
      float acc = 0.f;
      const float* W = p.out_w + i * 128 * 64 + t;
      for (int j = 0; j < 128; ++j) acc += yv[j] * W[j * 64];
      v[t] += acc;
    }
    __syncthreads();
  }
  if (t < 64) tf_out[blockIdx.x * 64 + t] = v[t];
}

// ---------------------------------------------------------------------------
// Kernel 2: head. rain gate per batch, fused(64x128) @ head_w1(128x64) via
// V_WMMA_F32_16X16X4_F32 (16 waves = 4x4 tiles of 16x16, 32 K-steps), BN+relu,
// final matvec + sigmoid. 1 block, 512 threads (16 wave32).
// ---------------------------------------------------------------------------
__global__ void __launch_bounds__(512)
head_kernel(const float* __restrict__ x, const float* __restrict__ tf,
            const float* __restrict__ logit_w, const float* __restrict__ logit_b,
            const float* __restrict__ gate_w, const float* __restrict__ gate_b,
            const float* __restrict__ head_w1, const float* __restrict__ head_b1,
            const float* __restrict__ bn_g, const float* __restrict__ bn_b,
            const float* __restrict__ bn_rm, const float* __restrict__ bn_rv,
            const float* __restrict__ head_w2, const float* __restrict__ head_b2,
            float* __restrict__ out)
{
  __shared__ float fused[64][132]; // padded vs 128 to dodge bank conflicts
  __shared__ float hs[64][66];
  __shared__ float rp[64];

  const int t = threadIdx.x;

  // rain_prob per batch: x[b, 0, 0:4] @ logit_w + logit_b -> sigmoid
  if (t < 64) {
    float acc = logit_b[0];
    for (int r = 0; r < 4; ++r) acc += x[t * 7168 + r] * logit_w[r];
    rp[t] = sigm_f(acc);
  }
  __syncthreads();

  // fused[b][e] = tf[e] * (rp[b]*gate_w[e] + gate_b[e])
  for (int idx = t; idx < 64 * 128; idx += 512) {
    const int b = idx >> 7, e = idx & 127;
    fused[b][e] = tf[e] * (rp[b] * gate_w[e] + gate_b[e]);
  }
  __syncthreads();

  // GEMM h = fused @ head_w1 with f32 WMMA 16x16x4
  const int wave = t >> 5, lane = t & 31;
  const int mt = wave >> 2, nt = wave & 3;
  const int half = lane >> 4, l16 = lane & 15;

  v8f c = {};
  const int m = mt * 16 + l16;       // A row for this lane
  const int n = nt * 16 + l16;       // B col for this lane
  for (int k0 = 0; k0 < 128; k0 += 4) {
    const int ka = k0 + half * 2;    // lanes 0-15: K{0,1}; lanes 16-31: K{2,3}
    v2f a; a.x = fused[m][ka]; a.y = fused[m][ka + 1];
    v2f b; b.x = head_w1[ka * 64 + n]; b.y = head_w1[(ka + 1) * 64 + n];
    c = __builtin_amdgcn_wmma_f32_16x16x4_f32(
        /*neg_a=*/false, a, /*neg_b=*/false, b,
        /*c_mod=*/(short)0, c, /*reuse_a=*/false, /*reuse_b=*/false);
  }

  // C/D layout: VGPR r -> M = r + 8*half, N = l16 (within tile). BN + relu.
  for (int r = 0; r < 8; ++r) {
    const int mo = mt * 16 + r + half * 8;
    const int no = nt * 16 + l16;
    float h = c[r] + head_b1[no];
    h = (h - bn_rm[no]) * rsqrtf(bn_rv[no] + BN_EPSF) * bn_g[no] + bn_b[no];
    hs[mo][no] = fmaxf(h, 0.f);
  }
  __syncthreads();

  if (t < 64) {
    float acc = head_b2[0];
    for (int nn = 0; nn < 64; ++nn) acc += hs[t][nn] * head_w2[nn];
    out[t] = sigm_f(acc);
  }
}

// ---------------------------------------------------------------------------
// Host launcher
// ---------------------------------------------------------------------------
extern "C" void kernel_launch(void* const* d_in, const int* in_sizes, int n_in,
                              void* d_out, int out_size, void* d_ws, size_t ws_size,
                              hipStream_t stream) {
  (void)n_in; (void)out_size; (void)ws_size;

  // Index maps for the two plausible flattening conventions.
  int I_x, I_ctx, I_pos;
  int T_lng, T_lnb, T_inw, T_cw, T_cb, T_xw, T_dtw, T_dtb, T_D, T_ow;
  int F_lng, F_lnb, F_inw, F_cw, F_cb, F_xw, F_dtw, F_dtb, F_D, F_ow;
  int I_lw, I_lb, I_gw, I_gb, I_hw1, I_hb1, I_bng, I_bnb, I_bnrm, I_bnrv, I_hw2, I_hb2;

  if (in_sizes[0] == 64 * 1024 * 7) {
    // setup_inputs() insertion order
    I_x = 0; I_ctx = 1; I_pos = 2;
    // topo: proj_w=3, proj_b=4, ln_g=5, ln_b=6, in_w=7, conv_w=8, conv_b=9,
    //       xproj_w=10, dt_w=11, dt_b=12, A_log=13, D=14, out_w=15
    T_lng = 5; T_lnb = 6; T_inw = 7; T_cw = 8; T_cb = 9; T_xw = 10;
    T_dtw = 11; T_dtb = 12; T_D = 14; T_ow = 15;
    F_lng = 18; F_lnb = 19; F_inw = 20; F_cw = 21; F_cb = 22; F_xw = 23;
    F_dtw = 24; F_dtb = 25; F_D = 27; F_ow = 28;
    I_lw = 29; I_lb = 30; I_gw = 31; I_gb = 32; I_hw1 = 33; I_hb1 = 34;
    I_bng = 35; I_bnb = 36; I_bnrm = 37; I_bnrv = 38; I_hw2 = 39; I_hb2 = 40;
  } else {
    // jax tree_leaves alphabetical order (params-dict sorted, 'params' < 'x')
    I_bnb = 0; I_bng = 1; I_bnrm = 2; I_bnrv = 3; I_ctx = 4;
    // fuel: A_log=5, D=6, conv_b=7, conv_w=8, dt_b=9, dt_w=10, in_w=11,
    //       ln_b=12, ln_g=13, out_w=14, proj_b=15, proj_w=16, xproj_w=17
    F_D = 6; F_cb = 7; F_cw = 8; F_dtb = 9; F_dtw = 10; F_inw = 11;
    F_lnb = 12; F_lng = 13; F_ow = 14; F_xw = 17;
    I_gb = 18; I_gw = 19; I_hb1 = 20; I_hb2 = 21; I_hw1 = 22; I_hw2 = 23;
    I_lb = 24; I_lw = 25; I_pos = 26;
    T_D = 28; T_cb = 29; T_cw = 30; T_dtb = 31; T_dtw = 32; T_inw = 33;
    T_lnb = 34; T_lng = 35; T_ow = 36; T_xw = 39;
    I_x = 40;
  }

  const float* F = nullptr; (void)F;
  #define FP(i) ((const float*)d_in[(i)])

  BranchP pt; BranchP pf;
  pt.ln_g = FP(T_lng); pt.ln_b = FP(T_lnb); pt.in_w = FP(T_inw);
  pt.conv_w = FP(T_cw); pt.conv_b = FP(T_cb); pt.xproj_w = FP(T_xw);
  pt.dt_w = FP(T_dtw); pt.dt_b = FP(T_dtb); pt.Dp = FP(T_D); pt.out_w = FP(T_ow);
  pf.ln_g = FP(F_lng); pf.ln_b = FP(F_lnb); pf.in_w = FP(F_inw);
  pf.conv_w = FP(F_cw); pf.conv_b = FP(F_cb); pf.xproj_w = FP(F_xw);
  pf.dt_w = FP(F_dtw); pf.dt_b = FP(F_dtb); pf.Dp = FP(F_D); pf.out_w = FP(F_ow);

  float* tf = (float*)d_ws; // 128 floats: [topo(64) | fuel(64)]

  branch_kernel<<<2, 256, 0, stream>>>(FP(I_ctx), FP(I_pos), pt, pf, tf);

  head_kernel<<<1, 512, 0, stream>>>(
      FP(I_x), tf,
      FP(I_lw), FP(I_lb), FP(I_gw), FP(I_gb),
      FP(I_hw1), FP(I_hb1), FP(I_bng), FP(I_bnb),
      FP(I_bnrm), FP(I_bnrv), FP(I_hw2), FP(I_hb2),
      (float*)d_out);

  #undef FP
}